// TaylorExp_13503377178735
// MI455X (gfx1250) — compile-verified
//
#include <hip/hip_runtime.h>
#include <hip/hip_bf16.h>

typedef __attribute__((ext_vector_type(2))) float v2f;
typedef __attribute__((ext_vector_type(8))) float v8f;

// One wave (32 lanes) handles one length-16 row:
//   out[0]        = 1
//   out[1..16]    = x * 0.5                      (1/sqrt(sqrt(16)))
//   out[17..272]  = (x_i * x_j) / (sqrt(2)*4)    via V_WMMA_F32_16X16X4_F32
__global__ void TaylorExp_kernel(const float* __restrict__ x,
                                 float* __restrict__ out,
                                 int nrows) {
    const int lane = threadIdx.x & 31;
    const int wave = threadIdx.x >> 5;
    const int row  = blockIdx.x * (blockDim.x >> 5) + wave;
    if (row >= nrows) return;          // wave-uniform: EXEC stays all-1 for WMMA

    const float* __restrict__ xr = x + (size_t)row * 16;

    // Lanes 0-15 fetch x[0..15]; lanes 16-31 carry zeros (they hold K=2/3 of A
    // and K>=1 rows of B, which must be zero for the rank-1 outer product).
    float xv = 0.0f;
    if (lane < 16) xv = xr[lane];

    // A: 16x4 f32, lanes 0-15 = M, VGPR0 = K=0 column -> x[M]; everything else 0.
    // B:  4x16 f32, lanes 0-15 = N, VGPR0 = K=0 row    -> x[N]; everything else 0.
    v2f a; a[0] = xv; a[1] = 0.0f;
    v2f b; b[0] = xv; b[1] = 0.0f;
    v8f c = {};

    // D = A x B + 0 = x . x^T  (16x16 f32 accumulator, 8 VGPRs)
    c = __builtin_amdgcn_wmma_f32_16x16x4_f32(
        /*neg_a=*/false, a, /*neg_b=*/false, b,
        /*c_mod=*/(short)0, c, /*reuse_a=*/false, /*reuse_b=*/false);

    float* __restrict__ o = out + (size_t)row * 273;

    // term1 + term2: lane 0 writes out[0] = 1; lanes 0-15 write out[1+lane] = x[lane]*0.5
    if (lane == 0) {
        __builtin_nontemporal_store(1.0f, o);
    }
    if (lane < 16) {
        __builtin_nontemporal_store(xv * 0.5f, o + 1 + lane);
    }

    // term3: D layout -> VGPR r: lanes 0-15 hold (M=r,   N=lane),
    //                    lanes 16-31 hold (M=r+8, N=lane-16)
    // out offset = 17 + M*16 + N ; each store = two contiguous 64B spans.
    const float s = 0.17677669529663687f;  // 1/(4*sqrt(2))
    const int   N = lane & 15;
    const int   mbase = (lane < 16) ? 0 : 8;
    float* __restrict__ oq = o + 17 + N;
#pragma unroll
    for (int r = 0; r < 8; ++r) {
        __builtin_nontemporal_store(c[r] * s, oq + (mbase + r) * 16);
    }
}

extern "C" void kernel_launch(void* const* d_in, const int* in_sizes, int n_in,
                              void* d_out, int out_size, void* d_ws, size_t ws_size,
                              hipStream_t stream) {
    const float* x   = (const float*)d_in[0];
    float*       out = (float*)d_out;

    const int nrows = in_sizes[0] / 16;        // 4*16*4096 = 262144 rows
    const int wavesPerBlock = 8;               // 256 threads = 8 wave32
    const int blocks = (nrows + wavesPerBlock - 1) / wavesPerBlock;

    TaylorExp_kernel<<<blocks, 256, 0, stream>>>(x, out, nrows);
}